// StockHeteGAT_24361054503615
// MI455X (gfx1250) — compile-verified
//
#include <hip/hip_runtime.h>
#include <math.h>

// Problem constants (match reference)
#define TSTEPS 20
#define NIN    6
#define HH     64
#define G3     192     // 3*H
#define HEADS_ 8
#define FDIM   8
#define CAP    3008    // >= N, nonzero list capacity per row

typedef float v2f __attribute__((ext_vector_type(2)));
typedef float v8f __attribute__((ext_vector_type(8)));

__device__ __forceinline__ float sigmoidf_(float x) { return 1.f / (1.f + __expf(-x)); }
__device__ __forceinline__ float leakyf_(float x)   { return x >= 0.f ? x : 0.2f * x; }

// ---------------------------------------------------------------------------
// GRU: one wave (32 lanes) per node, 8 nodes per 256-thread block.
// Lane handles hidden units k=lane and k=lane+32 -> all three gates for those
// units live in this lane; only h itself round-trips through LDS.
// Whh stored transposed [k][g] so lane accesses (stride 32 in g) hit distinct
// LDS banks. 20 sequential steps, h resident in LDS.
// ---------------------------------------------------------------------------
__global__ __launch_bounds__(256) void gru_kernel(
    const float* __restrict__ x,    // [N, 20, 6]
    const float* __restrict__ Wih,  // [192, 6]
    const float* __restrict__ Whh,  // [192, 64]
    const float* __restrict__ bih,  // [192]
    const float* __restrict__ bhh,  // [192]
    float* __restrict__ support,    // [N, 64]
    int N)
{
    __shared__ float sWhhT[HH * G3];    // [k][g]  48 KB
    __shared__ float sWihT[NIN * G3];   // [k][g]
    __shared__ float sbih[G3], sbhh[G3];
    __shared__ float sh[8][HH];
    __shared__ float sx[8][TSTEPS * NIN];

    const int tid  = threadIdx.x;
    const int wave = tid >> 5;
    const int lane = tid & 31;

    for (int i = tid; i < G3 * HH; i += 256) {
        int g = i / HH, k = i % HH;
        sWhhT[k * G3 + g] = Whh[i];
    }
    for (int i = tid; i < G3 * NIN; i += 256) {
        int g = i / NIN, k = i % NIN;
        sWihT[k * G3 + g] = Wih[i];
    }
    for (int i = tid; i < G3; i += 256) { sbih[i] = bih[i]; sbhh[i] = bhh[i]; }

    int n  = blockIdx.x * 8 + wave;
    int nn = (n < N) ? n : (N - 1);
    for (int i = lane; i < TSTEPS * NIN; i += 32) sx[wave][i] = x[(size_t)nn * (TSTEPS * NIN) + i];
    sh[wave][lane] = 0.f;
    sh[wave][lane + 32] = 0.f;
    __syncthreads();

    const int k0 = lane, k1 = lane + 32;
    for (int t = 0; t < TSTEPS; ++t) {
        // gh = h @ Whh^T + bhh  (6 gate outputs per lane: r/z/n for k0 and k1)
        float a0 = sbhh[k0],       a1 = sbhh[64 + k0],  a2 = sbhh[128 + k0];
        float a3 = sbhh[k1],       a4 = sbhh[64 + k1],  a5 = sbhh[128 + k1];
        for (int k = 0; k < HH; ++k) {
            float hk = sh[wave][k];
            const float* wr = &sWhhT[k * G3];
            a0 += hk * wr[k0];        a1 += hk * wr[64 + k0];  a2 += hk * wr[128 + k0];
            a3 += hk * wr[k1];        a4 += hk * wr[64 + k1];  a5 += hk * wr[128 + k1];
        }
        // gi = x_t @ Wih^T + bih
        float b0 = sbih[k0],       b1 = sbih[64 + k0],  b2 = sbih[128 + k0];
        float b3 = sbih[k1],       b4 = sbih[64 + k1],  b5 = sbih[128 + k1];
        for (int k = 0; k < NIN; ++k) {
            float xv = sx[wave][t * NIN + k];
            const float* wr = &sWihT[k * G3];
            b0 += xv * wr[k0];        b1 += xv * wr[64 + k0];  b2 += xv * wr[128 + k0];
            b3 += xv * wr[k1];        b4 += xv * wr[64 + k1];  b5 += xv * wr[128 + k1];
        }
        __syncthreads();   // everyone done reading sh
        float h0 = sh[wave][k0], h1 = sh[wave][k1];
        float r0 = sigmoidf_(b0 + a0), z0 = sigmoidf_(b1 + a1);
        float n0 = tanhf(b2 + r0 * a2);
        float r1 = sigmoidf_(b3 + a3), z1 = sigmoidf_(b4 + a4);
        float n1 = tanhf(b5 + r1 * a5);
        sh[wave][k0] = (1.f - z0) * n0 + z0 * h0;
        sh[wave][k1] = (1.f - z1) * n1 + z1 * h1;
        __syncthreads();
    }
    if (n < N) {
        support[(size_t)n * HH + k0] = sh[wave][k0];
        support[(size_t)n * HH + k1] = sh[wave][k1];
    }
}

// ---------------------------------------------------------------------------
// Generic f32 WMMA GEMM:  C[M,64] = A[M,64] @ B[64,64] (+add)(+bias1)(+bias2)
// One wave per 16x16 output tile; 16 x V_WMMA_F32_16X16X4_F32 over K=64.
// Fragment layouts per CDNA5 ISA 7.12.2 (32-bit A 16x4, B/C row-striped).
// ---------------------------------------------------------------------------
__global__ __launch_bounds__(32) void gemm64_wmma(
    const float* __restrict__ A, const float* __restrict__ B,
    const float* __restrict__ addv,   // nullable, [M,64]
    const float* __restrict__ bias1,  // nullable, [64]
    const float* __restrict__ bias2,  // nullable, [64]
    float* __restrict__ C, int M)
{
    const int m0   = blockIdx.x * 16;
    const int n0   = blockIdx.y * 16;
    const int lane = threadIdx.x;
    const int half = lane >> 4;       // 0: lanes 0-15, 1: lanes 16-31
    const int l    = lane & 15;

    int rowA = m0 + l;
    if (rowA >= M) rowA = M - 1;      // clamp; stores are guarded below

    v8f acc = {};
    for (int kk = 0; kk < 64; kk += 4) {
        v2f a, b;
        // A 16x4: vgpr0 = K(k0+2*half), vgpr1 = K(k0+2*half+1), M = l
        a.x = A[(size_t)rowA * 64 + kk + 2 * half];
        a.y = A[(size_t)rowA * 64 + kk + 2 * half + 1];
        // B 4x16: vgpr0 row K=k0+2*half striped over N=n0+l
        b.x = B[(size_t)(kk + 2 * half) * 64 + n0 + l];
        b.y = B[(size_t)(kk + 2 * half + 1) * 64 + n0 + l];
        acc = __builtin_amdgcn_wmma_f32_16x16x4_f32(
            false, a, false, b, (short)0, acc, false, false);
    }
    // C/D: vgpr r -> (M = m0 + r + 8*half, N = n0 + l)
    for (int r = 0; r < 8; ++r) {
        int row = m0 + r + 8 * half;
        int col = n0 + l;
        if (row < M) {
            float v = acc[r];
            if (addv)  v += addv[(size_t)row * 64 + col];
            if (bias1) v += bias1[col];
            if (bias2) v += bias2[col];
            C[(size_t)row * 64 + col] = v;
        }
    }
}

// ---------------------------------------------------------------------------
// f1[h,n] = sum_f s[n,h*8+f]*Wu[h,f];  f2 likewise with Wv.
// ---------------------------------------------------------------------------
__global__ __launch_bounds__(256) void f1f2_kernel(
    const float* __restrict__ s, const float* __restrict__ Wu,
    const float* __restrict__ Wv, float* __restrict__ f1,
    float* __restrict__ f2, int N)
{
    int t = blockIdx.x * 256 + threadIdx.x;
    if (t >= N * HEADS_) return;
    int n = t >> 3, h = t & 7;
    float u = 0.f, v = 0.f;
    for (int f = 0; f < FDIM; ++f) {
        float sv = s[(size_t)n * HH + h * FDIM + f];
        u += sv * Wu[h * FDIM + f];
        v += sv * Wv[h * FDIM + f];
    }
    f1[(size_t)h * N + n] = u;
    f2[(size_t)h * N + n] = v;
}

// ---------------------------------------------------------------------------
// GAT attention row kernel. One 256-thread block per destination node i.
// Phase A: stream adj row, deterministic ballot/prefix-sum compaction of
//          nonzeros into LDS (list sorted by j => deterministic FP order).
// Phase B: per-head (wave w = head w) max and sum-of-exp over nonzeros.
// Phase C: out[i, h*8+f] = sum_j softmax_j * s[j, h*8+f], 4-way k split.
// Masked softmax over the full row is exactly softmax over nonzeros
// (exp(-1e30 - max) == 0), zero-neighbor rows produce zeros.
// ---------------------------------------------------------------------------
__global__ __launch_bounds__(256) void gat_attn_kernel(
    const float* __restrict__ adj,  // [N,N]
    const float* __restrict__ s,    // [N,64]  per-head features h*8+f
    const float* __restrict__ f1,   // [8*N]  column term
    const float* __restrict__ f2,   // [8*N]  row term
    float* __restrict__ out,        // [N,64] raw attention output
    int N)
{
    __shared__ int   sidx[CAP];
    __shared__ float sval[CAP];
    __shared__ int   swcnt[8];
    __shared__ int   stot;
    __shared__ float smax[8], ssum[8];
    __shared__ float sacc[4][64];

    const int i    = blockIdx.x;
    const int tid  = threadIdx.x;
    const int wave = tid >> 5;
    const int lane = tid & 31;
    const float* __restrict__ arow = adj + (size_t)i * N;

    if (tid == 0) stot = 0;
    __syncthreads();

    for (int base = 0; base < N; base += 256) {
        int j = base + tid;
        if (base + 256 + tid < N)
            __builtin_prefetch(arow + base + 256 + tid, 0, 1);  // global_prefetch_b8
        float a = (j < N) ? arow[j] : 0.f;
        bool  p = (a != 0.f);
        unsigned m = (unsigned)__ballot(p);           // wave32: low 32 bits
        int prefix = __popc(m & ((1u << lane) - 1u));
        if (lane == 0) swcnt[wave] = __popc(m);
        __syncthreads();
        int off = stot;
        for (int w2 = 0; w2 < wave; ++w2) off += swcnt[w2];
        if (p) {
            int pos = off + prefix;
            if (pos < CAP) { sidx[pos] = j; sval[pos] = a; }
        }
        __syncthreads();
        if (tid == 0) {
            int tt = 0;
            for (int w2 = 0; w2 < 8; ++w2) tt += swcnt[w2];
            stot += tt;
        }
        __syncthreads();
    }

    int cnt = stot;
    if (cnt > CAP) cnt = CAP;
    if (cnt == 0) {
        for (int c = tid; c < HH; c += 256) out[(size_t)i * HH + c] = 0.f;
        return;
    }

    // Phase B: head = wave. Max, then sum of exp (butterfly reductions).
    {
        float fi = f2[(size_t)wave * N + i];
        float mx = -1e30f;
        for (int k = lane; k < cnt; k += 32) {
            int j = sidx[k];
            float sc = leakyf_(f1[(size_t)wave * N + j] + fi) * sval[k];
            mx = fmaxf(mx, sc);
        }
        for (int o = 16; o; o >>= 1) mx = fmaxf(mx, __shfl_xor(mx, o, 32));
        if (lane == 0) smax[wave] = mx;
        __syncthreads();
        float mm = smax[wave];
        float se = 0.f;
        for (int k = lane; k < cnt; k += 32) {
            int j = sidx[k];
            float sc = leakyf_(f1[(size_t)wave * N + j] + fi) * sval[k];
            se += __expf(sc - mm);
        }
        for (int o = 16; o; o >>= 1) se += __shfl_xor(se, o, 32);
        if (lane == 0) ssum[wave] = se;
        __syncthreads();
    }

    // Phase C: c = h*8+f output element, 4-way split over the neighbor list.
    {
        const int c   = tid & 63;
        const int seg = tid >> 6;
        const int h   = c >> 3;
        float mh  = smax[h];
        float inv = 1.f / ssum[h];
        float fih = f2[(size_t)h * N + i];
        float acc = 0.f;
        for (int k = seg; k < cnt; k += 4) {
            int j = sidx[k];
            float sc = leakyf_(f1[(size_t)h * N + j] + fih) * sval[k];
            acc += __expf(sc - mh) * s[(size_t)j * HH + c];
        }
        sacc[seg][c] = acc;
        __syncthreads();
        if (seg == 0) {
            float v = (sacc[0][c] + sacc[1][c] + sacc[2][c] + sacc[3][c]) * inv;
            out[(size_t)i * HH + c] = v;
        }
    }
}

// ---------------------------------------------------------------------------
// Semantic attention: per node, beta = softmax(tanh(E@W1+b1)@W2), emb = beta.E
// ---------------------------------------------------------------------------
__global__ __launch_bounds__(64) void sem_kernel(
    const float* __restrict__ s_self, const float* __restrict__ s_pos,
    const float* __restrict__ s_neg,  const float* __restrict__ W1,
    const float* __restrict__ b1,     const float* __restrict__ W2,
    float* __restrict__ emb, int N)
{
    const int n = blockIdx.x, f = threadIdx.x;
    __shared__ float ls[3][64];
    __shared__ float red[3][64];
    __shared__ float wv[3];
    ls[0][f] = s_self[(size_t)n * HH + f];
    ls[1][f] = s_pos [(size_t)n * HH + f];
    ls[2][f] = s_neg [(size_t)n * HH + f];
    __syncthreads();
    float t[3];
    for (int c = 0; c < 3; ++c) {
        float acc = b1[f];
        for (int k = 0; k < HH; ++k) acc += ls[c][k] * W1[k * HH + f];
        t[c] = tanhf(acc);
    }
    for (int c = 0; c < 3; ++c) red[c][f] = t[c] * W2[f];
    __syncthreads();
    for (int o = 32; o >= 1; o >>= 1) {
        if (f < o) { red[0][f] += red[0][f+o]; red[1][f] += red[1][f+o]; red[2][f] += red[2][f+o]; }
        __syncthreads();
    }
    if (f == 0) {
        float w0 = red[0][0], w1 = red[1][0], w2 = red[2][0];
        float mx = fmaxf(w0, fmaxf(w1, w2));
        float e0 = __expf(w0 - mx), e1 = __expf(w1 - mx), e2 = __expf(w2 - mx);
        float si = 1.f / (e0 + e1 + e2);
        wv[0] = e0 * si; wv[1] = e1 * si; wv[2] = e2 * si;
    }
    __syncthreads();
    emb[(size_t)n * HH + f] = wv[0]*ls[0][f] + wv[1]*ls[1][f] + wv[2]*ls[2][f];
}

// Column means of emb (single block, deterministic 4-way split per column).
__global__ __launch_bounds__(256) void colmean_kernel(
    const float* __restrict__ emb, float* __restrict__ mean, int N)
{
    __shared__ float part[256];
    int tid = threadIdx.x, c = tid & 63, seg = tid >> 6;
    float s = 0.f;
    for (int r = seg; r < N; r += 4) s += emb[(size_t)r * HH + c];
    part[tid] = s;
    __syncthreads();
    if (seg == 0) mean[c] = (part[c] + part[64+c] + part[128+c] + part[192+c]) / (float)N;
}

// PairNorm-SI + sigmoid(x @ predW + predb)
__global__ __launch_bounds__(64) void final_kernel(
    const float* __restrict__ emb, const float* __restrict__ mean,
    const float* __restrict__ predW, const float* __restrict__ predb,
    float* __restrict__ outp, int N)
{
    const int n = blockIdx.x, f = threadIdx.x;
    __shared__ float red[64];
    __shared__ float rn;
    float x = emb[(size_t)n * HH + f] - mean[f];
    red[f] = x * x;
    __syncthreads();
    for (int o = 32; o >= 1; o >>= 1) { if (f < o) red[f] += red[f + o]; __syncthreads(); }
    if (f == 0) rn = sqrtf(1e-6f + red[0]);
    __syncthreads();
    float xn = x / rn;
    red[f] = xn * predW[f];
    __syncthreads();
    for (int o = 32; o >= 1; o >>= 1) { if (f < o) red[f] += red[f + o]; __syncthreads(); }
    if (f == 0) outp[n] = 1.f / (1.f + __expf(-(red[0] + predb[0])));
}

// ---------------------------------------------------------------------------
extern "C" void kernel_launch(void* const* d_in, const int* in_sizes, int n_in,
                              void* d_out, int out_size, void* d_ws, size_t ws_size,
                              hipStream_t stream)
{
    const float* x         = (const float*)d_in[0];
    const float* pos_adj   = (const float*)d_in[1];
    const float* neg_adj   = (const float*)d_in[2];
    const float* gWih      = (const float*)d_in[3];
    const float* gWhh      = (const float*)d_in[4];
    const float* gbih      = (const float*)d_in[5];
    const float* gbhh      = (const float*)d_in[6];
    const float* pos_W     = (const float*)d_in[7];
    const float* pos_Wu    = (const float*)d_in[8];
    const float* pos_Wv    = (const float*)d_in[9];
    const float* pos_b     = (const float*)d_in[10];
    const float* pos_projW = (const float*)d_in[11];
    const float* pos_projb = (const float*)d_in[12];
    const float* neg_W     = (const float*)d_in[13];
    const float* neg_Wu    = (const float*)d_in[14];
    const float* neg_Wv    = (const float*)d_in[15];
    const float* neg_b     = (const float*)d_in[16];
    const float* neg_projW = (const float*)d_in[17];
    const float* neg_projb = (const float*)d_in[18];
    const float* self_W    = (const float*)d_in[19];
    const float* self_b    = (const float*)d_in[20];
    const float* mlpp_W    = (const float*)d_in[21];
    const float* mlpp_b    = (const float*)d_in[22];
    const float* mlpn_W    = (const float*)d_in[23];
    const float* mlpn_b    = (const float*)d_in[24];
    const float* sem_W1    = (const float*)d_in[25];
    const float* sem_b1    = (const float*)d_in[26];
    const float* sem_W2    = (const float*)d_in[27];
    const float* pred_W    = (const float*)d_in[28];
    const float* pred_b    = (const float*)d_in[29];

    const int N = in_sizes[0] / (TSTEPS * NIN);   // 3000
    float* out = (float*)d_out;

    // workspace layout
    float* ws = (float*)d_ws;
    size_t o = 0;
    float* support = ws + o; o += (size_t)N * HH;
    float* pos_sup = ws + o; o += (size_t)N * HH;
    float* neg_sup = ws + o; o += (size_t)N * HH;
    float* f1p     = ws + o; o += (size_t)HEADS_ * N;
    float* f2p     = ws + o; o += (size_t)HEADS_ * N;
    float* f1n     = ws + o; o += (size_t)HEADS_ * N;
    float* f2n     = ws + o; o += (size_t)HEADS_ * N;
    float* pos_gat = ws + o; o += (size_t)N * HH;
    float* neg_gat = ws + o; o += (size_t)N * HH;
    float* pos_s   = ws + o; o += (size_t)N * HH;
    float* neg_s   = ws + o; o += (size_t)N * HH;
    float* s_self  = ws + o; o += (size_t)N * HH;
    float* s_pos   = ws + o; o += (size_t)N * HH;
    float* s_neg   = ws + o; o += (size_t)N * HH;
    float* emb     = ws + o; o += (size_t)N * HH;
    float* cmean   = ws + o; o += 64;
    (void)o; (void)ws_size; (void)n_in; (void)out_size;

    const dim3 ggrid((N + 15) / 16, 4);

    // 1) GRU -> support
    gru_kernel<<<dim3((N + 7) / 8), dim3(256), 0, stream>>>(
        x, gWih, gWhh, gbih, gbhh, support, N);

    // 2) Per-head feature projections (WMMA GEMM)
    gemm64_wmma<<<ggrid, 32, 0, stream>>>(support, pos_W, nullptr, nullptr, nullptr, pos_sup, N);
    gemm64_wmma<<<ggrid, 32, 0, stream>>>(support, neg_W, nullptr, nullptr, nullptr, neg_sup, N);

    // 3) Attention logits terms
    int f12blocks = (N * HEADS_ + 255) / 256;
    f1f2_kernel<<<f12blocks, 256, 0, stream>>>(pos_sup, pos_Wu, pos_Wv, f1p, f2p, N);
    f1f2_kernel<<<f12blocks, 256, 0, stream>>>(neg_sup, neg_Wu, neg_Wv, f1n, f2n, N);

    // 4) Sparse-compacted masked softmax attention (HBM-bound adj streaming)
    gat_attn_kernel<<<N, 256, 0, stream>>>(pos_adj, pos_sup, f1p, f2p, pos_gat, N);
    gat_attn_kernel<<<N, 256, 0, stream>>>(neg_adj, neg_sup, f1n, f2n, neg_gat, N);

    // 5) GAT epilogue: out + b + support@projW + projb (WMMA GEMM fused epilogue)
    gemm64_wmma<<<ggrid, 32, 0, stream>>>(support, pos_projW, pos_gat, pos_b, pos_projb, pos_s, N);
    gemm64_wmma<<<ggrid, 32, 0, stream>>>(support, neg_projW, neg_gat, neg_b, neg_projb, neg_s, N);

    // 6) Channel embeddings (WMMA GEMMs)
    gemm64_wmma<<<ggrid, 32, 0, stream>>>(support, self_W, nullptr, self_b, nullptr, s_self, N);
    gemm64_wmma<<<ggrid, 32, 0, stream>>>(pos_s,   mlpp_W, nullptr, mlpp_b, nullptr, s_pos,  N);
    gemm64_wmma<<<ggrid, 32, 0, stream>>>(neg_s,   mlpn_W, nullptr, mlpn_b, nullptr, s_neg,  N);

    // 7) Semantic attention combine
    sem_kernel<<<N, 64, 0, stream>>>(s_self, s_pos, s_neg, sem_W1, sem_b1, sem_W2, emb, N);

    // 8) PairNorm-SI + prediction
    colmean_kernel<<<1, 256, 0, stream>>>(emb, cmean, N);
    final_kernel<<<N, 64, 0, stream>>>(emb, cmean, pred_W, pred_b, out, N);
}